// Sp_GCN_8383776162610
// MI455X (gfx1250) — compile-verified
//
#include <hip/hip_runtime.h>

// Sp-GCN on MI455X (gfx1250).
// Layer: relu(Ahat @ (X @ W)) twice. Fully fp32 (matches reference numerics)
// using V_WMMA_F32_16X16X4_F32 for the dense GEMMs. SpMM is edge-parallel
// atomic scatter (hardware global_atomic_add_f32); gathered feature matrix
// (51.2 MB) is L2-resident on the 192 MB L2.
//
// Workspace layout (requires 2 * N * 128 * 4 bytes = 102.4 MB):
//   ws0: Y   = (X|H) @ W     (GEMM output / SpMM gather source)
//   ws1: S0  = Ahat @ Y0     (layer-0 accumulator; relu fused into GEMM-2 A-load)
//   d_out:   layer-1 accumulator, relu'd in place.

typedef __attribute__((ext_vector_type(2))) float v2f;
typedef __attribute__((ext_vector_type(8))) float v8f;

#define FDIM 128

__global__ __launch_bounds__(256) void zero_f32(float* __restrict__ p, int n) {
  int i = blockIdx.x * blockDim.x + threadIdx.x;
  if (i < n) p[i] = 0.0f;
}

__global__ __launch_bounds__(256) void relu_f32(float* __restrict__ p, int n) {
  int i = blockIdx.x * blockDim.x + threadIdx.x;
  if (i < n) p[i] = fmaxf(p[i], 0.0f);
}

// Y[n_rows x 128] = act(A[n_rows x 128]) @ W[128 x 128]
// act = relu if relu_a, else identity.
// Block = 256 threads = 8 waves; each wave computes a 16x128 strip.
__global__ __launch_bounds__(256) void gemm128_wmma(const float* __restrict__ A,
                                                    const float* __restrict__ W,
                                                    float* __restrict__ Y,
                                                    int n_rows, int relu_a) {
  __shared__ float lw[FDIM * FDIM];  // 64 KB; WGP has 320 KB LDS

  // Cooperative 128x128 W load into LDS, float4-vectorized.
  {
    const float4* src = (const float4*)W;
    float4* dst = (float4*)lw;
    const int nvec = FDIM * FDIM / 4;  // 4096
    for (int i = threadIdx.x; i < nvec; i += 256) dst[i] = src[i];
  }
  __syncthreads();

  const int lane = threadIdx.x & 31;
  const int wave = threadIdx.x >> 5;
  const int m = lane & 15;             // M (for A) / N (for B,D) within tile
  const int kk = (lane >> 4) << 1;     // K sub-offset: lanes 0-15 -> 0, 16-31 -> 2
  const int row0 = blockIdx.x * 128 + wave * 16;

  // Clamp A row for the tail block: WMMA needs EXEC all ones, so we compute
  // garbage rows but never store them.
  int arow = row0 + m;
  if (arow >= n_rows) arow = n_rows - 1;
  const float* ap = A + (size_t)arow * FDIM;

  v8f acc[8];
#pragma unroll
  for (int t = 0; t < 8; ++t) acc[t] = (v8f)0.0f;

  for (int k = 0; k < FDIM; k += 4) {
    // A fragment, 16x4 f32: VGPR0 = K=kk, VGPR1 = K=kk+1 (ISA 7.12.2)
    float a0 = ap[k + kk];
    float a1 = ap[k + kk + 1];
    if (relu_a) {
      a0 = fmaxf(a0, 0.0f);
      a1 = fmaxf(a1, 0.0f);
    }
    v2f a;
    a.x = a0;
    a.y = a1;
#pragma unroll
    for (int t = 0; t < 8; ++t) {
      // B fragment, 4x16 f32: row K striped across lanes within a VGPR.
      v2f b;
      b.x = lw[(k + kk) * FDIM + t * 16 + m];
      b.y = lw[(k + kk + 1) * FDIM + t * 16 + m];
      acc[t] = __builtin_amdgcn_wmma_f32_16x16x4_f32(
          false, a, false, b, (short)0, acc[t], false, false);
    }
  }

  // D layout: VGPR r -> row (r + 8*(lane>=16)), col = lane%16 (ISA 7.12.2)
  const int rbase = row0 + ((lane >> 4) << 3);
#pragma unroll
  for (int t = 0; t < 8; ++t) {
#pragma unroll
    for (int r = 0; r < 8; ++r) {
      int row = rbase + r;
      if (row < n_rows) Y[(size_t)row * FDIM + t * 16 + m] = acc[t][r];
    }
  }
}

// S[rows[e]] += vals[e] * Y[cols[e]]  for every edge, one wave per edge.
// Each lane covers 4 consecutive features (float4 load + 4 f32 atomics).
__global__ __launch_bounds__(256) void spmm_scatter(const int* __restrict__ erows,
                                                    const int* __restrict__ ecols,
                                                    const float* __restrict__ evals,
                                                    const float* __restrict__ Y,
                                                    float* __restrict__ S,
                                                    int nedges) {
  int gid = blockIdx.x * blockDim.x + threadIdx.x;
  int e = gid >> 5;
  int lane = gid & 31;
  if (e >= nedges) return;

  int r = erows[e];
  int c = ecols[e];
  float v = evals[e];

  const float4* yp = (const float4*)(Y + (size_t)c * FDIM);
  float4 y = yp[lane];  // features [4*lane, 4*lane+4)

  float* out = S + (size_t)r * FDIM + lane * 4;
  unsafeAtomicAdd(out + 0, v * y.x);
  unsafeAtomicAdd(out + 1, v * y.y);
  unsafeAtomicAdd(out + 2, v * y.z);
  unsafeAtomicAdd(out + 3, v * y.w);
}

extern "C" void kernel_launch(void* const* d_in, const int* in_sizes, int n_in,
                              void* d_out, int out_size, void* d_ws, size_t ws_size,
                              hipStream_t stream) {
  const float* X    = (const float*)d_in[0];  // [N,128]
  const float* vals = (const float*)d_in[1];  // [E]
  const float* W0   = (const float*)d_in[2];  // [128,128]
  const float* W1   = (const float*)d_in[3];  // [128,128]
  const int* erows  = (const int*)d_in[4];    // [E]
  const int* ecols  = (const int*)d_in[5];    // [E]
  float* out = (float*)d_out;                 // [N,128]

  const int nnodes = in_sizes[0] / FDIM;
  const int nedges = in_sizes[1];
  const int nelem = nnodes * FDIM;

  float* Ybuf = (float*)d_ws;                        // N*128 fp32
  float* Sbuf = Ybuf + (size_t)nnodes * FDIM;        // N*128 fp32

  const int gz = (nelem + 255) / 256;
  const int ggemm = (nnodes + 127) / 128;
  const long long sp_threads = (long long)nedges * 32;
  const int gspmm = (int)((sp_threads + 255) / 256);

  // ---- Layer 0: S0 = Ahat @ (X @ W0) ----
  zero_f32<<<gz, 256, 0, stream>>>(Sbuf, nelem);
  gemm128_wmma<<<ggemm, 256, 0, stream>>>(X, W0, Ybuf, nnodes, 0);
  spmm_scatter<<<gspmm, 256, 0, stream>>>(erows, ecols, vals, Ybuf, Sbuf, nedges);

  // ---- Layer 1: out = relu(Ahat @ (relu(S0) @ W1)) ----
  zero_f32<<<gz, 256, 0, stream>>>(out, nelem);
  gemm128_wmma<<<ggemm, 256, 0, stream>>>(Sbuf, W1, Ybuf, nnodes, 1);  // relu fused into A-load
  spmm_scatter<<<gspmm, 256, 0, stream>>>(erows, ecols, vals, Ybuf, out, nedges);
  relu_f32<<<gz, 256, 0, stream>>>(out, nelem);
}